// GraphNet_35064113004880
// MI455X (gfx1250) — compile-verified
//
#include <hip/hip_runtime.h>
#include <math.h>

typedef float v2f __attribute__((ext_vector_type(2)));
typedef float v8f __attribute__((ext_vector_type(8)));

#define Nn 8192
#define Dd 512
#define EPSC 2.220446049250313e-16f
#define MAXNBR 64
#define COLSPLIT 8

#define WMMA_F32(a, b, c) \
    __builtin_amdgcn_wmma_f32_16x16x4_f32(false, (a), false, (b), (short)0, (c), false, false)

// ---------------------------------------------------------------------------
// Generic f32 WMMA GEMM: C[M x Nc] = A[M x K] @ B[K x Nc]  (+bias, +relu)
// Compile-time K/LD* -> immediate-offset global loads, no 64-bit addr chains.
// block = 256 threads (8 waves); wave -> 16x16 tile; grid = (M/128, Nc/16)
// ---------------------------------------------------------------------------
template <int K, int LDA, int LDB, int LDC, int EPI>
__global__ void gemm_wmma_kernel(const float* __restrict__ A,
                                 const float* __restrict__ B,
                                 const float* __restrict__ bias,
                                 float* __restrict__ C) {
    const int lane = threadIdx.x & 31;
    const int wv   = threadIdx.x >> 5;
    const int l15  = lane & 15;
    const int hi   = lane >> 4;                 // 0 or 1
    const int rowBase = blockIdx.x * 128 + wv * 16;
    const int col0    = blockIdx.y * 16;
    const float* aBase = A + (size_t)(rowBase + l15) * LDA + 2 * hi;
    const float* bBase = B + (size_t)(2 * hi) * LDB + col0 + l15;

    v8f acc = {};
#pragma unroll 4
    for (int k = 0; k < K; k += 4) {
        v2f a = *(const v2f*)(aBase + k);
        v2f b;
        b.x = bBase[k * LDB];
        b.y = bBase[(k + 1) * LDB];
        acc = WMMA_F32(a, b, acc);
    }

    const int cRow0 = rowBase + 8 * hi;
    const int col   = col0 + l15;
    float bv = (EPI == 1) ? bias[col] : 0.0f;
#pragma unroll
    for (int r = 0; r < 8; ++r) {
        float v = acc[r] + bv;
        if (EPI == 1) v = fmaxf(v, 0.0f);
        C[(size_t)(cRow0 + r) * LDC + col] = v;
    }
}

// ---------------------------------------------------------------------------
// RelationNet tail: h1(relu, precomputed) -> h2(relu) -> sigma -> emb, sq
// block = 64 threads, grid = N
// ---------------------------------------------------------------------------
__global__ void relnet_tail_kernel(const float* __restrict__ x,
                                   const float* __restrict__ h1,
                                   const float* __restrict__ w2,   // [64][16]
                                   const float* __restrict__ b2,
                                   const float* __restrict__ w3,   // [16]
                                   const float* __restrict__ b3,
                                   float* __restrict__ emb,
                                   float* __restrict__ sq) {
    __shared__ float sh1[64];
    __shared__ float sh2[16];
    __shared__ float red[64];
    __shared__ float sinv;
    const int n = blockIdx.x;
    const int t = threadIdx.x;

    sh1[t] = h1[(size_t)n * 64 + t];
    __syncthreads();
    if (t < 16) {
        float a = b2[t];
        for (int k = 0; k < 64; ++k) a += sh1[k] * w2[k * 16 + t];
        sh2[t] = fmaxf(a, 0.0f);
    }
    __syncthreads();
    if (t == 0) {
        float s = b3[0];
        for (int k = 0; k < 16; ++k) s += sh2[k] * w3[k];
        sinv = 1.0f / (s + EPSC);
    }
    __syncthreads();
    const float inv = sinv;
    float local = 0.0f;
    for (int c = t; c < Dd; c += 64) {
        float e = x[(size_t)n * Dd + c] * inv;
        emb[(size_t)n * Dd + c] = e;
        local += e * e;
    }
    red[t] = local;
    __syncthreads();
    for (int s = 32; s > 0; s >>= 1) {
        if (t < s) red[t] += red[t + s];
        __syncthreads();
    }
    if (t == 0) sq[n] = red[0];
}

// ---------------------------------------------------------------------------
// Fused Gram + top-3: score(i,j) = 2*emb_i.emb_j - sq_j (monotone in Wg).
// Register-blocked: wave computes 16 rows x 64 cols (4 accumulators) per
// step, sharing one A fragment -> 1.25 b64 loads per WMMA (L2-resident emb).
// block = 128 (4 waves, 16 rows each); grid = (N/64, COLSPLIT).
// Partial per-split top-3 candidates written out; no atomics (deterministic).
// ---------------------------------------------------------------------------
__global__ void gram_topk_kernel(const float* __restrict__ emb,
                                 const float* __restrict__ sq,
                                 float* __restrict__ cand_val,
                                 int* __restrict__ cand_idx) {
    __shared__ float Ss[4][16][65];
    const int lane = threadIdx.x & 31;
    const int wv   = threadIdx.x >> 5;
    const int l15  = lane & 15;
    const int hi   = lane >> 4;
    const int row0 = blockIdx.x * 64 + wv * 16;
    const int colBeg = blockIdx.y * (Nn / COLSPLIT);
    const int colEnd = colBeg + (Nn / COLSPLIT);
    const float* aBase = emb + (size_t)(row0 + l15) * Dd + 2 * hi;

    float v0 = -3.4e38f, v1 = -3.4e38f, v2 = -3.4e38f;
    int   i0 = 0x7fffffff, i1 = 0x7fffffff, i2 = 0x7fffffff;

    for (int j0 = colBeg; j0 < colEnd; j0 += 64) {
        const float* bB0 = emb + (size_t)(j0 +  0 + l15) * Dd + 2 * hi;
        const float* bB1 = emb + (size_t)(j0 + 16 + l15) * Dd + 2 * hi;
        const float* bB2 = emb + (size_t)(j0 + 32 + l15) * Dd + 2 * hi;
        const float* bB3 = emb + (size_t)(j0 + 48 + l15) * Dd + 2 * hi;
        v8f acc0 = {}, acc1 = {}, acc2 = {}, acc3 = {};
#pragma unroll 4
        for (int k = 0; k < Dd; k += 4) {
            v2f a  = *(const v2f*)(aBase + k);
            v2f b0 = *(const v2f*)(bB0 + k);
            v2f b1 = *(const v2f*)(bB1 + k);
            v2f b2 = *(const v2f*)(bB2 + k);
            v2f b3 = *(const v2f*)(bB3 + k);
            acc0 = WMMA_F32(a, b0, acc0);
            acc1 = WMMA_F32(a, b1, acc1);
            acc2 = WMMA_F32(a, b2, acc2);
            acc3 = WMMA_F32(a, b3, acc3);
        }
        const float sq0 = sq[j0 +  0 + l15];
        const float sq1 = sq[j0 + 16 + l15];
        const float sq2 = sq[j0 + 32 + l15];
        const float sq3 = sq[j0 + 48 + l15];
#pragma unroll
        for (int r = 0; r < 8; ++r) {
            const int sr = r + 8 * hi;
            Ss[wv][sr][ 0 + l15] = 2.0f * acc0[r] - sq0;
            Ss[wv][sr][16 + l15] = 2.0f * acc1[r] - sq1;
            Ss[wv][sr][32 + l15] = 2.0f * acc2[r] - sq2;
            Ss[wv][sr][48 + l15] = 2.0f * acc3[r] - sq3;
        }
        __syncthreads();
        // lane pair (l, l+16) owns row l; lane covers cols [32*hi, 32*hi+32)
        for (int c = 0; c < 32; ++c) {
            const int col = 32 * hi + c;
            const float s = Ss[wv][l15][col];
            const int j = j0 + col;
            if (s > v0)      { v2=v1;i2=i1; v1=v0;i1=i0; v0=s;i0=j; }
            else if (s > v1) { v2=v1;i2=i1; v1=s;i1=j; }
            else if (s > v2) { v2=s;i2=j; }
        }
        __syncthreads();
    }

    // merge lane pair -> per-wave top3 for this column split
    const int src = (lane + 16) & 31;
    float u0 = __shfl(v0, src, 32), u1 = __shfl(v1, src, 32), u2 = __shfl(v2, src, 32);
    int   j0c = __shfl(i0, src, 32), j1c = __shfl(i1, src, 32), j2c = __shfl(i2, src, 32);
    if (lane < 16) {
        float av[6] = { v0, v1, v2, u0, u1, u2 };
        int   ai[6] = { i0, i1, i2, j0c, j1c, j2c };
        const int row = row0 + lane;
        float* cv = cand_val + ((size_t)blockIdx.y * Nn + row) * 3;
        int*   ci = cand_idx + ((size_t)blockIdx.y * Nn + row) * 3;
#pragma unroll
        for (int t = 0; t < 3; ++t) {
            int best = t;
            for (int m = t + 1; m < 6; ++m)
                if (av[m] > av[best] || (av[m] == av[best] && ai[m] < ai[best])) best = m;
            float tv = av[best]; av[best] = av[t]; av[t] = tv;
            int   ti = ai[best]; ai[best] = ai[t]; ai[t] = ti;
            cv[t] = av[t];
            ci[t] = ai[t];
        }
    }
}

// merge COLSPLIT partial top3 lists per row -> final idx3
__global__ void topk_merge_kernel(const float* __restrict__ cand_val,
                                  const int* __restrict__ cand_idx,
                                  int* __restrict__ idx3) {
    const int row = blockIdx.x * blockDim.x + threadIdx.x;
    if (row >= Nn) return;
    int p0 = -1, p1 = -1;
    for (int t = 0; t < 3; ++t) {
        float bv = -3.4e38f; int bi = 0x7fffffff;
        for (int s = 0; s < COLSPLIT; ++s) {
            const float* cv = cand_val + ((size_t)s * Nn + row) * 3;
            const int*   ci = cand_idx + ((size_t)s * Nn + row) * 3;
            for (int k = 0; k < 3; ++k) {
                const int c = ci[k];
                if (c == p0 || c == p1) continue;
                const float v = cv[k];
                if (v > bv || (v == bv && c < bi)) { bv = v; bi = c; }
            }
        }
        idx3[row * 3 + t] = bi;
        if (t == 0) p0 = bi; else if (t == 1) p1 = bi;
    }
}

// ---------------------------------------------------------------------------
// Deterministic adjacency build: A[i][j] = j in top3(i) || i in top3(j)
// one wave per row; ballot+popcount compaction (ascending j, no atomics)
// ---------------------------------------------------------------------------
__global__ void build_nbr_kernel(const int* __restrict__ idx3,
                                 int* __restrict__ nbr,
                                 int* __restrict__ ncnt,
                                 float* __restrict__ dinv) {
    const int lane = threadIdx.x & 31;
    const int wv   = threadIdx.x >> 5;
    const int row  = blockIdx.x * 8 + wv;
    const int t0 = idx3[row * 3 + 0];
    const int t1 = idx3[row * 3 + 1];
    const int t2 = idx3[row * 3 + 2];
    int cnt = 0;
    for (int jb = 0; jb < Nn; jb += 32) {
        const int j = jb + lane;
        const int a0 = idx3[j * 3 + 0];
        const int a1 = idx3[j * 3 + 1];
        const int a2 = idx3[j * 3 + 2];
        const bool p = (j == t0) | (j == t1) | (j == t2) |
                       (a0 == row) | (a1 == row) | (a2 == row);
        const unsigned m = (unsigned)__ballot(p);
        const int pos = cnt + __popc(m & ((1u << lane) - 1u));
        if (p && pos < MAXNBR) nbr[(size_t)row * MAXNBR + pos] = j;
        cnt += __popc(m);
    }
    if (lane == 0) {
        ncnt[row] = cnt < MAXNBR ? cnt : MAXNBR;
        dinv[row] = rsqrtf((float)(cnt > 0 ? cnt : 1));
    }
}

// ---------------------------------------------------------------------------
// Sparse normalized aggregation + bias + ELU (+ optional residual)
// out_i = resid_i + elu(dinv_i * sum_j dinv_j * M_j + bias)
// block per node
// ---------------------------------------------------------------------------
__global__ void agg_kernel(const float* __restrict__ M,
                           const float* __restrict__ bias,
                           const float* __restrict__ resid,
                           const int* __restrict__ nbr,
                           const int* __restrict__ ncnt,
                           const float* __restrict__ dinv,
                           float* __restrict__ out, int C) {
    __shared__ int   sj[MAXNBR];
    __shared__ float sd[MAXNBR];
    const int i = blockIdx.x;
    const int cnt = ncnt[i];
    if (threadIdx.x < cnt) {
        const int j = nbr[(size_t)i * MAXNBR + threadIdx.x];
        sj[threadIdx.x] = j;
        sd[threadIdx.x] = dinv[j];
    }
    __syncthreads();
    const float di = dinv[i];
    for (int c = threadIdx.x; c < C; c += blockDim.x) {
        float acc = 0.0f;
        for (int s = 0; s < cnt; ++s)
            acc += sd[s] * M[(size_t)sj[s] * C + c];
        float v = di * acc + bias[c];
        v = v > 0.0f ? v : (expf(v) - 1.0f);      // elu
        if (resid) v += resid[(size_t)i * C + c];
        out[(size_t)i * C + c] = v;
    }
}

// ---------------------------------------------------------------------------
extern "C" void kernel_launch(void* const* d_in, const int* in_sizes, int n_in,
                              void* d_out, int out_size, void* d_ws, size_t ws_size,
                              hipStream_t stream) {
    const float* x      = (const float*)d_in[0];
    const float* rel_w1 = (const float*)d_in[1];
    const float* rel_b1 = (const float*)d_in[2];
    const float* rel_w2 = (const float*)d_in[3];
    const float* rel_b2 = (const float*)d_in[4];
    const float* rel_w3 = (const float*)d_in[5];
    const float* rel_b3 = (const float*)d_in[6];
    const float* gcn_w1 = (const float*)d_in[7];
    const float* gcn_b1 = (const float*)d_in[8];
    const float* gcn_w2 = (const float*)d_in[9];
    const float* gcn_b2 = (const float*)d_in[10];
    const float* mlp_w1 = (const float*)d_in[11];
    const float* mlp_b1 = (const float*)d_in[12];
    const float* mlp_w2 = (const float*)d_in[13];
    float* out = (float*)d_out;

    float* ws = (float*)d_ws;
    size_t off = 0;
    auto alloc = [&](size_t n) { float* p = ws + off; off += (n + 63) & ~(size_t)63; return p; };
    float* emb      = alloc((size_t)Nn * Dd);
    float* sq       = alloc(Nn);
    float* h_rel1   = alloc((size_t)Nn * 64);
    float* cand_val = alloc((size_t)COLSPLIT * Nn * 3);
    int*   cand_idx = (int*)alloc((size_t)COLSPLIT * Nn * 3);
    int*   idx3     = (int*)alloc((size_t)Nn * 3);
    int*   nbr      = (int*)alloc((size_t)Nn * MAXNBR);
    int*   ncnt     = (int*)alloc(Nn);
    float* dinv     = alloc(Nn);
    float* xw1      = alloc((size_t)Nn * 128);
    float* h1g      = alloc((size_t)Nn * 128);
    float* h1w2     = alloc((size_t)Nn * Dd);
    float* hg       = alloc((size_t)Nn * Dd);
    float* t1       = alloc((size_t)Nn * 128);

    // 1) relation net layer 1: relu(x @ rel_w1 + b1)  [8192 x 64]
    gemm_wmma_kernel<512, 512, 64, 64, 1>
        <<<dim3(Nn / 128, 64 / 16), 256, 0, stream>>>(x, rel_w1, rel_b1, h_rel1);
    // 2) relnet tail -> emb, sq
    relnet_tail_kernel<<<Nn, 64, 0, stream>>>(x, h_rel1, rel_w2, rel_b2,
                                              rel_w3, rel_b3, emb, sq);
    // 3) fused Gram + per-split top3 (WMMA-dominant kernel)
    gram_topk_kernel<<<dim3(Nn / 64, COLSPLIT), 128, 0, stream>>>(
        emb, sq, cand_val, cand_idx);
    // 4) merge splits -> final top3 indices
    topk_merge_kernel<<<(Nn + 255) / 256, 256, 0, stream>>>(cand_val, cand_idx, idx3);
    // 5) deterministic symmetric adjacency + dinv
    build_nbr_kernel<<<Nn / 8, 256, 0, stream>>>(idx3, nbr, ncnt, dinv);
    // 6) GCN layer 1
    gemm_wmma_kernel<512, 512, 128, 128, 0>
        <<<dim3(Nn / 128, 128 / 16), 256, 0, stream>>>(x, gcn_w1, nullptr, xw1);
    agg_kernel<<<Nn, 128, 0, stream>>>(xw1, gcn_b1, nullptr, nbr, ncnt, dinv, h1g, 128);
    // 7) GCN layer 2 + residual
    gemm_wmma_kernel<128, 128, 512, 512, 0>
        <<<dim3(Nn / 128, Dd / 16), 256, 0, stream>>>(h1g, gcn_w2, nullptr, h1w2);
    agg_kernel<<<Nn, 256, 0, stream>>>(h1w2, gcn_b2, x, nbr, ncnt, dinv, hg, Dd);
    // 8) MLP head
    gemm_wmma_kernel<512, 512, 128, 128, 1>
        <<<dim3(Nn / 128, 128 / 16), 256, 0, stream>>>(hg, mlp_w1, mlp_b1, t1);
    gemm_wmma_kernel<128, 128, 512, 512, 0>
        <<<dim3(Nn / 128, Dd / 16), 256, 0, stream>>>(t1, mlp_w2, nullptr, out);
}